// AUC_86990267613596
// MI455X (gfx1250) — compile-verified
//
#include <hip/hip_runtime.h>
#include <stdint.h>

// ---------------- problem constants (from reference) ----------------
#define N_ROWS   500000
#define N_COLS   64
#define STEPN    30
#define NBINS    31           // STEP + 1 segments
#define WAVESZ   32           // CDNA5 wave32
#define WAVES    8            // waves per block
#define THREADS  (WAVES * WAVESZ)
#define BLOCKS   1024
#define DEPTH    4            // async pipeline buffers (3 loads in flight)
#define NSTAGES  ((N_ROWS * N_COLS) / (4 * WAVESZ))   // 250000 full stages

// ---------------- typed pointers for the async builtin ----------------
typedef int v4i __attribute__((ext_vector_type(4)));
typedef __attribute__((address_space(1))) v4i g1_v4i;   // global
typedef __attribute__((address_space(3))) v4i l3_v4i;   // LDS

// Async global -> LDS 128-bit copy (one instruction per wave moves 512 B).
// Tracked with ASYNCcnt; in-flight data lives in LDS, not VGPRs.
__device__ __forceinline__ void async_copy_b128(const void* gsrc, void* ldst) {
#if __has_builtin(__builtin_amdgcn_global_load_async_to_lds_b128)
  v4i* g0 = (v4i*)gsrc;          // drop const / retype in generic AS
  v4i* l0 = (v4i*)ldst;
  __builtin_amdgcn_global_load_async_to_lds_b128(
      (g1_v4i*)g0, (l3_v4i*)l0, /*offset=*/0, /*cpol=*/0);
#else
  unsigned lds_off = (unsigned)(uintptr_t)ldst;   // low 32 bits of generic == LDS offset
  asm volatile("global_load_async_to_lds_b128 %0, %1, off"
               :: "v"(lds_off), "v"((unsigned long long)gsrc)
               : "memory");
#endif
}

__device__ __forceinline__ void wait_async_le3() {
#if __has_builtin(__builtin_amdgcn_s_wait_asynccnt)
  __builtin_amdgcn_s_wait_asynccnt(DEPTH - 1);
#else
  asm volatile("s_wait_asynccnt 0x3" ::: "memory");
#endif
}

// bin = floor(30 * sigmoid(x)), clamped to [0,30]
__device__ __forceinline__ int bin_of(float x) {
  float e = __builtin_amdgcn_exp2f(x * -1.44269504088896340736f); // e^{-x}
  float s = __builtin_amdgcn_rcpf(1.0f + e);                      // sigmoid
  int b = (int)(s * 30.0f);                                      // trunc == floor (s >= 0)
  b = b < 0 ? 0 : b;
  b = b > STEPN ? STEPN : b;
  return b;
}

// ---------------- kernel 1: histogram (BW-bound main pass) ----------------
__global__ void __launch_bounds__(THREADS)
auc_hist_kernel(const float* __restrict__ out,
                const int*   __restrict__ tgt,
                unsigned*    __restrict__ gh) {
  // 16 KB staging (4-deep pipeline per wave) + 2 KB per-wave histograms
  __shared__ float4   stage[WAVES][DEPTH][WAVESZ];
  __shared__ unsigned hist[WAVES][2][32];

  const int tid  = threadIdx.x;
  const int lane = tid & (WAVESZ - 1);
  const int wave = tid >> 5;

  // zero per-wave histograms (512 words / 256 threads)
  ((unsigned*)hist)[tid]       = 0u;
  ((unsigned*)hist)[tid + 256] = 0u;
  __syncthreads();

  const float4* __restrict__ src = (const float4*)out;
  const int stride = gridDim.x * WAVES;
  int s = blockIdx.x * WAVES + wave;

  // Prime: keep DEPTH-1 loads in flight. Out-of-range stages load tile 0
  // (L2-hot dummy) so ASYNCcnt is a compile-time constant at every wait.
#pragma unroll
  for (int d = 0; d < DEPTH - 1; ++d) {
    const int sd = s + d * stride;
    const float4* g = (sd < NSTAGES) ? &src[(size_t)sd * WAVESZ + lane]
                                     : &src[lane];
    async_copy_b128(g, &stage[wave][d][lane]);
  }

  int buf = 0;
  while (s < NSTAGES) {
    // issue stage s + 3*stride into the buffer freed last iteration
    const int sp = s + (DEPTH - 1) * stride;
    const float4* g = (sp < NSTAGES) ? &src[(size_t)sp * WAVESZ + lane]
                                     : &src[lane];
    async_copy_b128(g, &stage[wave][(buf + DEPTH - 1) & (DEPTH - 1)][lane]);

    wait_async_le3();              // in-order completion -> stage s resident

    // process current tile from LDS (ds_load_b128)
    const int    i4 = s * WAVESZ + lane;     // float4 index, < 8M
    const float4 v  = stage[wave][buf][lane];
    const int    n  = i4 >> 4;               // 16 float4 per row (C=64)
    const int    c0 = (i4 & 15) << 2;
    const int    t  = __ldg(&tgt[n]);        // 16 lanes share n -> L1/L2 hit

    {
      int b = bin_of(v.x);
      atomicAdd(&hist[wave][0][b], 1u);
      if (c0 + 0 == t) atomicAdd(&hist[wave][1][b], 1u);
    }
    {
      int b = bin_of(v.y);
      atomicAdd(&hist[wave][0][b], 1u);
      if (c0 + 1 == t) atomicAdd(&hist[wave][1][b], 1u);
    }
    {
      int b = bin_of(v.z);
      atomicAdd(&hist[wave][0][b], 1u);
      if (c0 + 2 == t) atomicAdd(&hist[wave][1][b], 1u);
    }
    {
      int b = bin_of(v.w);
      atomicAdd(&hist[wave][0][b], 1u);
      if (c0 + 3 == t) atomicAdd(&hist[wave][1][b], 1u);
    }

    s += stride;
    buf = (buf + 1) & (DEPTH - 1);
  }

  __syncthreads();
  // reduce 8 per-wave histograms -> 2 global atomics per bin per block
  if (tid < 64) {
    const int k = tid >> 5;        // 0 = all, 1 = trues
    const int b = tid & 31;
    unsigned sum = 0;
    for (int w = 0; w < WAVES; ++w) sum += hist[w][k][b];
    if (b < NBINS && sum) atomicAdd(&gh[k * 32 + b], sum);
  }
}

// ---------------- kernel 0: zero the global histogram ----------------
__global__ void auc_zero_kernel(unsigned* __restrict__ gh) {
  gh[threadIdx.x] = 0u;   // 64 slots: [0..31]=hist_all, [32..63]=hist_true
}

// ---------------- kernel 2: tiny cumsum + trapezoid ----------------
__global__ void auc_finalize_kernel(const unsigned* __restrict__ gh,
                                    float* __restrict__ out) {
  if (threadIdx.x != 0) return;

  float ht[NBINS], ha[NBINS];
  float trues = 0.0f, total = 0.0f;
  for (int b = 0; b < NBINS; ++b) {
    ha[b] = (float)gh[b];
    ht[b] = (float)gh[32 + b];
    total += ha[b];
    trues += ht[b];
  }
  const float falses = total - trues;

  float tp_asc[STEPN], fp_asc[STEPN];
  float ct = 0.0f, cf = 0.0f;
  for (int k = 0; k < STEPN; ++k) {
    ct += ht[k];
    cf += (ha[k] - ht[k]);
    tp_asc[k] = trues - ct;
    fp_asc[k] = falses - cf;
  }

  const float eps = 1e-8f;
  float prev_tpr = 0.0f, prev_fpr = 0.0f, area = 0.0f;
  for (int i = 0; i < STEPN; ++i) {
    const float tpr = tp_asc[STEPN - 1 - i] / (trues + eps);
    const float fpr = fp_asc[STEPN - 1 - i] / (falses + eps);
    const float w    = fabsf(fpr - prev_fpr);
    const float tmin = fminf(tpr, prev_tpr);
    const float tmax = fmaxf(tpr, prev_tpr);
    area += w * tmin + 0.5f * w * (tmax - tmin);
    prev_tpr = tpr;
    prev_fpr = fpr;
  }
  out[0] = area;
}

// ---------------- launcher ----------------
extern "C" void kernel_launch(void* const* d_in, const int* in_sizes, int n_in,
                              void* d_out, int out_size, void* d_ws, size_t ws_size,
                              hipStream_t stream) {
  const float* output = (const float*)d_in[0];
  const int*   target = (const int*)d_in[1];
  unsigned*    gh     = (unsigned*)d_ws;      // 64 u32 slots
  float*       res    = (float*)d_out;

  auc_zero_kernel<<<1, 64, 0, stream>>>(gh);
  auc_hist_kernel<<<BLOCKS, THREADS, 0, stream>>>(output, target, gh);
  auc_finalize_kernel<<<1, 32, 0, stream>>>(gh, res);
}